// KNNLayer_71966472011987
// MI455X (gfx1250) — compile-verified
//
#include <hip/hip_runtime.h>

typedef float v2f __attribute__((ext_vector_type(2)));
typedef float v8f __attribute__((ext_vector_type(8)));

#define B_Q        2048
#define N_TR       65536
#define DIM        128
#define K_NN       16
#define NUM_CLS    12

#define QTILE      16
#define NQT        (B_Q / QTILE)     // 128 query tiles
#define CHUNK      1024              // n's per workgroup in pass 1
#define NCHUNKS    (N_TR / CHUNK)    // 64
#define WAVES      8                 // 256 threads / wave32
#define NPW        (CHUNK / WAVES)   // 128 n per wave
#define NSUB       (NPW / 16)        // 8 wmma n-tiles per wave
#define KSTEPS     (DIM / 4)         // 32 chained 16x16x4 steps
#define PF         4                 // B-operand software prefetch depth
#define SENT       3.4e38f

// ---------------------------------------------------------------------------
// Pass 0: r2[n] = ||X_train[n]||^2
// ---------------------------------------------------------------------------
__global__ void knn_r2_kernel(const float* __restrict__ Xtr,
                              float* __restrict__ r2) {
    int n = blockIdx.x * blockDim.x + threadIdx.x;
    const float4* p = (const float4*)(Xtr + (size_t)n * DIM);
    float s = 0.f;
#pragma unroll
    for (int i = 0; i < DIM / 4; ++i) {
        float4 v = p[i];
        s += v.x * v.x + v.y * v.y + v.z * v.z + v.w * v.w;
    }
    r2[n] = s;
}

// ---------------------------------------------------------------------------
// Pass 1: fp32 WMMA distance scores (r2 - 2*dot) + per-chunk top-16
//   grid = (NQT, NCHUNKS), block = 256 (8 waves)
//   Two independent accumulator tiles per wave (hide WMMA result latency),
//   depth-4 rotating prefetch of B operands (keep 8 loads outstanding).
// ---------------------------------------------------------------------------
__launch_bounds__(256, 1)
__global__ void knn_dist_topk_kernel(const float* __restrict__ Xte,
                                     const float* __restrict__ Xtr,
                                     const float* __restrict__ r2,
                                     float* __restrict__ wsD,
                                     int*   __restrict__ wsI) {
    __shared__ float sdist[QTILE][CHUNK];   // 64 KB

    const int qt   = blockIdx.x;            // query tile
    const int ch   = blockIdx.y;            // n chunk
    const int lane = threadIdx.x & 31;
    const int wave = threadIdx.x >> 5;

    const int r     = lane & 15;            // row within 16 (M for A, N for B/C)
    const int khalf = (lane >> 4) * 2;      // K sub-offset per ISA 16x4 fp32 layout
    const int mrow  = qt * QTILE + r;

    // Preload A operand for full K=128.
    // Lane L holds A[M = L&15][K = 4*i + 2*(L>>4) .. +1] as a float2.
    v2f a[KSTEPS];
    const float* arow = Xte + (size_t)mrow * DIM;
#pragma unroll
    for (int i = 0; i < KSTEPS; ++i)
        a[i] = *(const v2f*)(arow + i * 4 + khalf);

    const int nWave = ch * CHUNK + wave * NPW;
    for (int t = 0; t < NSUB; t += 2) {
        const int nB0 = nWave + t * 16;
        const int nB1 = nB0 + 16;
        const float* brow0 = Xtr + (size_t)(nB0 + r) * DIM;  // B[k][n] = Xtr[n][k]
        const float* brow1 = Xtr + (size_t)(nB1 + r) * DIM;

        // Software-pipelined B fetches: rotating depth-PF buffers.
        v2f pb0[PF], pb1[PF];
#pragma unroll
        for (int i = 0; i < PF; ++i) {
            pb0[i] = *(const v2f*)(brow0 + i * 4 + khalf);
            pb1[i] = *(const v2f*)(brow1 + i * 4 + khalf);
        }

        v8f acc0 = {};
        v8f acc1 = {};
#pragma unroll
        for (int i = 0; i < KSTEPS; ++i) {
            const int slot = i % PF;
            v2f b0 = pb0[slot];
            v2f b1 = pb1[slot];
            if (i + PF < KSTEPS) {
                pb0[slot] = *(const v2f*)(brow0 + (i + PF) * 4 + khalf);
                pb1[slot] = *(const v2f*)(brow1 + (i + PF) * 4 + khalf);
            }
            acc0 = __builtin_amdgcn_wmma_f32_16x16x4_f32(
                false, a[i], false, b0, (short)0, acc0, false, false);
            acc1 = __builtin_amdgcn_wmma_f32_16x16x4_f32(
                false, a[i], false, b1, (short)0, acc1, false, false);
        }

        // C/D layout: lane holds column n = nB + (lane&15); VGPR j -> row j + 8*(lane>>4)
        const int   nloc0 = wave * NPW + t * 16 + r;
        const float rr0   = r2[nB0 + r];
        const float rr1   = r2[nB1 + r];
        const int   mhi   = (lane >> 4) * 8;
#pragma unroll
        for (int j = 0; j < 8; ++j) {
            sdist[mhi + j][nloc0]      = rr0 - 2.0f * acc0[j];
            sdist[mhi + j][nloc0 + 16] = rr1 - 2.0f * acc1[j];
        }
    }
    __syncthreads();

    // Per-chunk top-16 per query; wave w owns queries 2w, 2w+1.
    for (int qq = 0; qq < 2; ++qq) {
        const int m = wave * 2 + qq;
        float* row = &sdist[m][0];
        const int q = qt * QTILE + m;
        float* outD = wsD + ((size_t)q * NCHUNKS + ch) * K_NN;
        int*   outI = wsI + ((size_t)q * NCHUNKS + ch) * K_NN;

        for (int it = 0; it < K_NN; ++it) {
            float bv = SENT; int bi = -1;
            for (int jj = lane; jj < CHUNK; jj += 32) {
                float v = row[jj];
                if (v < bv) { bv = v; bi = jj; }
            }
#pragma unroll
            for (int off = 16; off >= 1; off >>= 1) {
                float ov = __shfl_xor(bv, off, 32);
                int   oi = __shfl_xor(bi, off, 32);
                if (ov < bv || (ov == bv && oi >= 0 && oi < bi)) { bv = ov; bi = oi; }
            }
            if (lane == 0) {
                outD[it] = bv;
                outI[it] = ch * CHUNK + bi;
                row[bi]  = SENT;          // knock out winner for next round
            }
        }
    }
}

// ---------------------------------------------------------------------------
// Pass 2: merge 64 chunk-lists -> global top-16, label vote. 1 wave / query.
// ---------------------------------------------------------------------------
__launch_bounds__(32, 1)
__global__ void knn_merge_vote_kernel(const float* __restrict__ wsD,
                                      const int*   __restrict__ wsI,
                                      const int*   __restrict__ ytr,
                                      float*       __restrict__ out) {
    __shared__ float cd[NCHUNKS * K_NN];   // 1024 candidates
    __shared__ int   ci[NCHUNKS * K_NN];
    __shared__ int   winner[K_NN];
    __shared__ int   labels[K_NN];

    const int q    = blockIdx.x;
    const int lane = threadIdx.x;
    const float* srcD = wsD + (size_t)q * NCHUNKS * K_NN;
    const int*   srcI = wsI + (size_t)q * NCHUNKS * K_NN;
    for (int j = lane; j < NCHUNKS * K_NN; j += 32) { cd[j] = srcD[j]; ci[j] = srcI[j]; }
    __syncthreads();

    for (int it = 0; it < K_NN; ++it) {
        float bv = SENT; int bidx = 0x7fffffff; int bj = -1;
        for (int j = lane; j < NCHUNKS * K_NN; j += 32) {
            float v = cd[j];
            int  id = ci[j];
            if (v < bv || (v == bv && id < bidx)) { bv = v; bidx = id; bj = j; }
        }
#pragma unroll
        for (int off = 16; off >= 1; off >>= 1) {
            float ov = __shfl_xor(bv, off, 32);
            int   oi = __shfl_xor(bidx, off, 32);
            int   oj = __shfl_xor(bj, off, 32);
            if (ov < bv || (ov == bv && oi < bidx)) { bv = ov; bidx = oi; bj = oj; }
        }
        if (lane == 0) { winner[it] = bidx; cd[bj] = SENT; }
    }
    __syncthreads();

    if (lane < K_NN) labels[lane] = ytr[winner[lane]];
    __syncthreads();

    if (lane < NUM_CLS) {
        int cnt = 0;
#pragma unroll
        for (int t = 0; t < K_NN; ++t) cnt += (labels[t] == lane) ? 1 : 0;
        out[(size_t)q * NUM_CLS + lane] = (float)cnt * (1.0f / (float)K_NN);
    }
}

// ---------------------------------------------------------------------------
extern "C" void kernel_launch(void* const* d_in, const int* in_sizes, int n_in,
                              void* d_out, int out_size, void* d_ws, size_t ws_size,
                              hipStream_t stream) {
    (void)in_sizes; (void)n_in; (void)out_size; (void)ws_size;

    const float* Xte = (const float*)d_in[0];   // [B, D] fp32
    const float* Xtr = (const float*)d_in[1];   // [N, D] fp32
    const int*   ytr = (const int*)d_in[2];     // [N] int32
    float* out = (float*)d_out;                 // [B, NUM_CLS] fp32

    // workspace layout: [wsD 8MB][wsI 8MB][r2 256KB]
    char* ws = (char*)d_ws;
    const size_t nCand = (size_t)B_Q * NCHUNKS * K_NN;   // 2,097,152
    float* wsD = (float*)ws;
    int*   wsI = (int*)(ws + nCand * sizeof(float));
    float* r2  = (float*)(ws + nCand * (sizeof(float) + sizeof(int)));

    knn_r2_kernel<<<N_TR / 256, 256, 0, stream>>>(Xtr, r2);

    dim3 g1(NQT, NCHUNKS);
    knn_dist_topk_kernel<<<g1, 256, 0, stream>>>(Xte, Xtr, r2, wsD, wsI);

    knn_merge_vote_kernel<<<B_Q, 32, 0, stream>>>(wsD, wsI, ytr, out);
}